// SConv2_2997887172767
// MI455X (gfx1250) — compile-verified
//
#include <hip/hip_runtime.h>

#define HH 256
#define WW 256

typedef __attribute__((ext_vector_type(2))) float v2f;
typedef __attribute__((ext_vector_type(8))) float v8f;

// ---- CDNA5 async-to-LDS path ----------------------------------------
// The builtin exists only in the device (gfx1250) pass; the host pass still
// parses kernel bodies, so stub the macros there. #error guards the device
// pass so a silent fallback is impossible.
#if defined(__HIP_DEVICE_COMPILE__)
#  if !defined(__has_builtin) || !__has_builtin(__builtin_amdgcn_global_load_async_to_lds_b32)
#    error "gfx1250 device pass missing __builtin_amdgcn_global_load_async_to_lds_b32"
#  endif
#  if __has_builtin(__builtin_amdgcn_s_wait_asynccnt)
#    define WAIT_ASYNC(n) __builtin_amdgcn_s_wait_asynccnt(n)
#  else
#    define WAIT_ASYNC(n) asm volatile("s_wait_asynccnt " #n ::: "memory")
#  endif
   // ROCm clang-22 prototype (from diagnostics): (AS1 int*, AS3 int*, imm, imm)
#  define ASYNC_ROW_B32(gp, lp)                                              \
     __builtin_amdgcn_global_load_async_to_lds_b32(                          \
         (__attribute__((address_space(1))) int*)(gp),                       \
         (__attribute__((address_space(3))) int*)(lp), 0, 0)
#else
#  define WAIT_ASYNC(n)      (void)0
#  define ASYNC_ROW_B32(gp, lp) (void)0
#endif

// =====================================================================
// Kernel A: kern[8,576] = leaky_relu(rep[8,512] @ w_key[576,512]^T)
// One wave per 16-wide N tile; K stepped by 4 via V_WMMA_F32_16X16X4_F32.
// A(16x4 f32) layout: lane = {M=lane&15}; VGPR0 holds K=khalf, VGPR1 K=khalf+1,
// khalf = 0 for lanes 0-15, 2 for lanes 16-31.  B(4x16) mirrors this.
// C(16x16): VGPR i, lanes 0-15 -> M=i (only M<8 valid / stored).
// Pad rows 8..15 of A read rep row 0 unmasked: WMMA rows are independent,
// so garbage only lands in C rows 8..15, which are never stored.
// =====================================================================
__global__ void __launch_bounds__(32) gen_kern_wmma(
    const float* __restrict__ rep,   // [8, 512]
    const float* __restrict__ wk,    // [576, 512]
    float* __restrict__ kern)        // [8, 576]
{
  const int tile  = blockIdx.x;          // 0..35
  const int lane  = threadIdx.x;         // 0..31
  const int m     = lane & 15;           // A row (batch), B column within tile
  const int khalf = (lane >> 4) << 1;    // 0 or 2
  const float* arow = rep + (m < 8 ? m : 0) * 512;
  const float* brow = wk + (tile * 16 + m) * 512;

  v8f acc = {};
  for (int k0 = 0; k0 < 512; k0 += 4) {
    v2f a = *(const v2f*)(arow + k0 + khalf);
    v2f b = *(const v2f*)(brow + k0 + khalf);
    acc = __builtin_amdgcn_wmma_f32_16x16x4_f32(
        false, a, false, b, (short)0, acc, false, false);
  }

  if (lane < 16) {
#pragma unroll
    for (int i = 0; i < 8; ++i) {                 // M = i (batch row)
      float v = acc[i];
      v = v > 0.0f ? v : 0.1f * v;                // fused LeakyReLU(0.1)
      kern[i * 576 + tile * 16 + lane] = v;
    }
  }
}

// =====================================================================
// Kernel B: dynamic depthwise 3x3 conv, reflection pad, fp32.
// Block = 288 threads (9 waves) per (b, c, 64-row strip).
// Threads 0..257 stage one padded row (258 floats) into a 6-slot LDS ring
// via async-to-LDS (depth-3 pipeline, s_wait_asynccnt 2 + 1 barrier/row).
// Threads 0..255 compute one output column: 9 LDS reads + 9 FMAs.
// =====================================================================
__device__ __forceinline__ int reflect255(int r) {
  r = (r < 0) ? -r : r;
  return (r > 255) ? (510 - r) : r;
}

__global__ void __launch_bounds__(288) dyn_dwconv(
    const float* __restrict__ x,     // [8, 64, 256, 256]
    const float* __restrict__ kern,  // [8, 576] == [8*64, 9]
    float* __restrict__ out)         // [8, 64, 256, 256]
{
  __shared__ float rows[6][264];     // ring of 6 padded rows (258 used)
  __shared__ float kws[9];

  const int tx    = threadIdx.x;
  const int strip = blockIdx.x;                       // 0..3
  const int bc    = blockIdx.z * 64 + blockIdx.y;     // b*64 + c
  const int h0    = strip * 64;

  if (tx < 9) kws[tx] = kern[bc * 9 + tx];

  const float* xp = x   + (size_t)bc * (HH * WW);
  float*       op = out + (size_t)bc * (HH * WW);

  const bool loader = (tx < 258);
  const int  scol   = reflect255(tx - 1);             // source column w/ reflect

  // Prologue: issue rows h0-1 .. h0+2 into slots 0..3.
#pragma unroll
  for (int i = 0; i < 4; ++i) {
    if (loader) {
      const int r = reflect255(h0 - 1 + i);
      const float* g = xp + r * WW + scol;
      ASYNC_ROW_B32(g, &rows[i][tx]);
    }
  }

  __syncthreads();                    // kws visible (async rows awaited in loop)
  float kk[9];
#pragma unroll
  for (int t = 0; t < 9; ++t) kk[t] = kws[t];

  for (int hh = 0; hh < 64; ++hh) {
    const int h = h0 + hh;

    // Issue row h+3 into slot (hh+4)%6; slot(r) = (r-(h0-1)) mod 6.
    if (loader) {
      const int r = reflect255(h + 3);                // tail rows just reflect
      const float* g = xp + r * WW + scol;
      ASYNC_ROW_B32(g, &rows[(hh + 4) % 6][tx]);
    }

    WAIT_ASYNC(2);                    // rows up to h+1 resident (per wave)
    __syncthreads();                  // all waves' portions resident

    if (tx < 256) {
      const float* r0 = rows[(hh + 0) % 6];           // row h-1 (padded)
      const float* r1 = rows[(hh + 1) % 6];           // row h
      const float* r2 = rows[(hh + 2) % 6];           // row h+1
      float acc = r0[tx]     * kk[0] + r0[tx + 1] * kk[1] + r0[tx + 2] * kk[2]
                + r1[tx]     * kk[3] + r1[tx + 1] * kk[4] + r1[tx + 2] * kk[5]
                + r2[tx]     * kk[6] + r2[tx + 1] * kk[7] + r2[tx + 2] * kk[8];
      op[h * WW + tx] = acc;
    }
  }
}

extern "C" void kernel_launch(void* const* d_in, const int* in_sizes, int n_in,
                              void* d_out, int out_size, void* d_ws, size_t ws_size,
                              hipStream_t stream) {
  (void)in_sizes; (void)n_in; (void)out_size; (void)ws_size;
  const float* x   = (const float*)d_in[0];   // [8,64,256,256]
  const float* rep = (const float*)d_in[1];   // [8,512]
  const float* wk  = (const float*)d_in[2];   // [576,512]
  float* out  = (float*)d_out;                // [8,64,256,256]
  float* kern = (float*)d_ws;                 // [8,576] scratch (18 KB)

  gen_kern_wmma<<<dim3(36), dim3(32), 0, stream>>>(rep, wk, kern);
  dyn_dwconv<<<dim3(4, 64, 8), dim3(288), 0, stream>>>(x, kern, out);
}